// filtered_correlation1D_58780922413132
// MI455X (gfx1250) — compile-verified
//
#include <hip/hip_runtime.h>
#include <hip/hip_bf16.h>

typedef __attribute__((ext_vector_type(2))) float v2f;
typedef __attribute__((ext_vector_type(8))) float v8f;

#define NB   2
#define CCH  32
#define HH   128
#define WW   256
#define DMAX 48
#define HW   (HH * WW)   // 32768 = 2^15

// ---------------------------------------------------------------------------
// Stage 1: banded Gram matrix  z[n,i,h,w] = sum_c x[n,c,h,w] * y[n,c,h,w-i]
// computed as 16x16 tiles of G = X^T * Y per (n,h) row with f32 WMMA (K=4).
// One wave32 per (n, h, w-tile). 4 diagonal-offset tiles cover the 48 band.
// ---------------------------------------------------------------------------
__global__ __launch_bounds__(256) void corr_band_wmma(
    const float* __restrict__ x, const float* __restrict__ y,
    float* __restrict__ z)
{
    const int lane = threadIdx.x & 31;
    const int tile = blockIdx.x * 8 + (threadIdx.x >> 5); // 4096 waves total
    const int wt   = tile & 15;          // 16 w-tiles per row
    const int h    = (tile >> 4) & (HH - 1);
    const int nb   = tile >> 11;
    const int w0   = wt << 4;

    const float* __restrict__ xp = x + nb * CCH * HW + h * WW; // xp[c*HW + w]
    const float* __restrict__ yp = y + nb * CCH * HW + h * WW;

    const int m    = lane & 15;   // A: M index / B: N index / C,D: N index
    const int half = lane >> 4;   // selects K pair (A/B) and M+8 (C/D)

#pragma unroll
    for (int t = 0; t < 4; ++t) {
        const int d0 = t << 4;        // diagonal offset of this G tile
        const int v0 = w0 - d0;       // y-column base (w' = w - i)
        v8f acc = {0.f, 0.f, 0.f, 0.f, 0.f, 0.f, 0.f, 0.f};

        // branch-free left-edge handling (keeps EXEC all-1s for WMMA)
        const int   wp  = v0 + m;
        const int   wpc = wp < 0 ? 0 : wp;
        const float bm  = wp < 0 ? 0.f : 1.f;

#pragma unroll
        for (int kk = 0; kk < 8; ++kk) {          // K = 32 in steps of 4
            const int c0 = (kk << 2) + (half << 1);
            v2f a, b;
            // A (16x4 f32): lane->M=m, VGPR v + 2*half -> K
            a.x = xp[c0 * HW + w0 + m];
            a.y = xp[(c0 + 1) * HW + w0 + m];
            // B (4x16 f32): lane->N=m, VGPR v + 2*half -> K
            b.x = yp[c0 * HW + wpc] * bm;
            b.y = yp[(c0 + 1) * HW + wpc] * bm;
            acc = __builtin_amdgcn_wmma_f32_16x16x4_f32(
                false, a, false, b, (short)0, acc, false, false);
        }

        // C/D layout: reg r, lane -> M = r + 8*half, N = lane&15
        // G[w0+M][v0+N] contributes z[i = d0+M-N][w = w0+M]
#pragma unroll
        for (int r = 0; r < 8; ++r) {
            const int mm   = r + (half << 3);
            const int disp = d0 + mm - m;
            if (disp >= 0 && disp < DMAX) {
                z[(nb * DMAX + disp) * HW + h * WW + (w0 + mm)] = acc[r];
            }
        }
    }
}

// ---------------------------------------------------------------------------
// Stage 2: per-pixel dynamic 5x5 dilated(2) filtering + bias.
// out[n,i,h,w] = bias[n,h,w] + sum_{kh,kw} z[n,i,h+2(kh-2),w+2(kw-2)]
//                                          * filters[n,kh*5+kw,h,w]
// ---------------------------------------------------------------------------
__global__ __launch_bounds__(256) void dynfilter_apply(
    const float* __restrict__ z, const float* __restrict__ filt,
    const float* __restrict__ bias, float* __restrict__ out)
{
    const int idx = blockIdx.x * blockDim.x + threadIdx.x;
    if (idx >= NB * DMAX * HW) return;

    const int w  = idx & (WW - 1);
    const int h  = (idx >> 8) & (HH - 1);
    const int di = (idx >> 15) % DMAX;
    const int nb = idx / (DMAX << 15);

    const float* __restrict__ zp = z + (nb * DMAX + di) * HW;
    const float* __restrict__ fp = filt + nb * 25 * HW + h * WW + w;

    float acc = bias[nb * HW + h * WW + w];
#pragma unroll
    for (int kh = 0; kh < 5; ++kh) {
        const int  hz  = h + 2 * (kh - 2);
        const bool hok = (unsigned)hz < (unsigned)HH;
#pragma unroll
        for (int kw = 0; kw < 5; ++kw) {
            const int  wz = w + 2 * (kw - 2);
            const bool ok = hok && ((unsigned)wz < (unsigned)WW);
            const float zv = ok ? zp[hz * WW + wz] : 0.f;
            acc = fmaf(zv, fp[(kh * 5 + kw) * HW], acc);
        }
    }
    out[idx] = acc;
}

// ---------------------------------------------------------------------------
extern "C" void kernel_launch(void* const* d_in, const int* in_sizes, int n_in,
                              void* d_out, int out_size, void* d_ws, size_t ws_size,
                              hipStream_t stream)
{
    const float* x    = (const float*)d_in[0]; // (2,32,128,256)
    const float* y    = (const float*)d_in[1]; // (2,32,128,256)
    const float* filt = (const float*)d_in[2]; // (2,25,128,256)
    const float* bias = (const float*)d_in[3]; // (2,1,128,256)
    float*       out  = (float*)d_out;         // (2,48,128,256)
    float*       z    = (float*)d_ws;          // needs NB*DMAX*H*W*4 = 12.6 MB

    // Stage 1: 4096 waves (one per n,h,w-tile), 8 waves per 256-thread block
    corr_band_wmma<<<NB * HH * (WW / 16) / 8, 256, 0, stream>>>(x, y, z);

    // Stage 2: one thread per output element
    const int total = NB * DMAX * HW;
    dynfilter_apply<<<(total + 255) / 256, 256, 0, stream>>>(z, filt, bias, out);
}